// GRUAutoencoder_6227702579468
// MI455X (gfx1250) — compile-verified
//
#include <hip/hip_runtime.h>
#include <math.h>

typedef __attribute__((ext_vector_type(2))) float v2f;
typedef __attribute__((ext_vector_type(8))) float v8f;

#define K_DIM 256
#define BATCH 32
#define SEQ   128
#define HID   256
#define G3    768     // 3*HID
#define TOK   4096    // BATCH*SEQ

__device__ __forceinline__ float sigmoidf_(float x) { return 1.0f / (1.0f + __expf(-x)); }

__device__ __forceinline__ v8f wmma4(v2f a, v2f b, v8f c) {
  return __builtin_amdgcn_wmma_f32_16x16x4_f32(false, a, false, b, (short)0, c, false, false);
}

// ---------------- embedding gather (padding_idx = 0) ----------------
__global__ void embed_kernel(const int* __restrict__ xs, const int* __restrict__ xc,
                             const float* __restrict__ es, const float* __restrict__ ec,
                             float* __restrict__ out) {
  int idx = blockIdx.x * blockDim.x + threadIdx.x;
  if (idx >= TOK * HID) return;
  int tok = idx >> 8, j = idx & 255;
  int s = xs[tok], c = xc[tok];
  float v = 0.f;
  if (s != 0) v += es[s * HID + j];
  if (c != 0) v += ec[c * HID + j];
  out[idx] = v;
}

// ---------------- combined = emb + encoded (broadcast over time) ----------------
__global__ void combine_kernel(const float* __restrict__ emb, const float* __restrict__ enc,
                               float* __restrict__ out) {
  int idx = blockIdx.x * blockDim.x + threadIdx.x;
  if (idx >= TOK * HID) return;
  int tok = idx >> 8;
  int b = tok >> 7;           // tok / SEQ
  int j = idx & 255;
  out[idx] = emb[idx] + enc[b * HID + j];
}

// ---------------- Y[M,N] = X[M,256] @ W[N,256]^T + bias[N]  (fp32 WMMA) ----------------
// One wave owns a 16 x 64 strip (4 adjacent 16x16 tiles, shared A fragment).
// K loop processes 16 elements per chunk: 20 batched b64 loads, then 16 WMMAs,
// so one partial loadcnt wait covers the whole batch (instead of wait-per-WMMA).
// NT=true streams the output with non-temporal stores (write-once logits must not
// evict the L2-resident weight matrix).
template <bool NT>
__global__ void gemm_wmma_f32(const float* __restrict__ X, const float* __restrict__ W,
                              const float* __restrict__ bias, float* __restrict__ Y,
                              int M, int N, int nGroups) {
  int wid = (blockIdx.x * blockDim.x + threadIdx.x) >> 5;
  int totalWaves = (M >> 4) * nGroups;
  if (wid >= totalWaves) return;           // uniform per wave -> EXEC all-ones for WMMA
  int tm = wid / nGroups;
  int g  = wid % nGroups;

  int lane  = threadIdx.x & 31;
  int mlo   = lane & 15;
  int khalf = (lane >> 4) << 1;            // 0 or 2
  int hi8   = (lane >> 4) << 3;            // 0 or 8

  const float* xrow = X + (tm * 16 + mlo) * K_DIM + khalf;
  const float* wrow[4];
  int ncol[4];
#pragma unroll
  for (int t = 0; t < 4; t++) {
    int n = (g * 4 + t) * 16 + mlo;
    ncol[t] = n;
    int nc = (n < N) ? n : (N - 1);        // clamp: garbage cols are never stored
    wrow[t] = W + nc * K_DIM + khalf;
    __builtin_prefetch(wrow[t], 0, 0);     // global_prefetch_b8 on the weight stream
  }
  __builtin_prefetch(xrow, 0, 0);

  v8f acc0 = {}, acc1 = {}, acc2 = {}, acc3 = {};
#pragma unroll 2
  for (int kc = 0; kc < K_DIM; kc += 16) {
    v2f a[4], b0[4], b1[4], b2[4], b3[4];
#pragma unroll
    for (int s = 0; s < 4; s++) {
      int k = kc + 4 * s;
      a[s].x  = xrow[k];     a[s].y  = xrow[k + 1];
      b0[s].x = wrow[0][k];  b0[s].y = wrow[0][k + 1];
      b1[s].x = wrow[1][k];  b1[s].y = wrow[1][k + 1];
      b2[s].x = wrow[2][k];  b2[s].y = wrow[2][k + 1];
      b3[s].x = wrow[3][k];  b3[s].y = wrow[3][k + 1];
    }
#pragma unroll
    for (int s = 0; s < 4; s++) {
      acc0 = wmma4(a[s], b0[s], acc0);
      acc1 = wmma4(a[s], b1[s], acc1);
      acc2 = wmma4(a[s], b2[s], acc2);
      acc3 = wmma4(a[s], b3[s], acc3);
    }
  }

  v8f accs[4] = {acc0, acc1, acc2, acc3};
#pragma unroll
  for (int t = 0; t < 4; t++) {
    int n = ncol[t];
    if (n < N) {
      float bv = bias[n];
#pragma unroll
      for (int i = 0; i < 8; i++) {
        int m = tm * 16 + i + hi8;
        float val = accs[t][i] + bv;
        float* p = Y + (size_t)m * N + n;
        if (NT) __builtin_nontemporal_store(val, p);
        else    *p = val;
      }
    }
  }
}

// ---------------- sequential GRU recurrence: one workgroup, h & gh in LDS ----------------
// gh = h @ W_hh^T computed with WMMA (M=32 -> 2 tile rows, N=768 -> 48 tile cols, 96 tiles / 32 waves)
__global__ void __launch_bounds__(1024)
gru_recur_kernel(const float* __restrict__ xW, const float* __restrict__ Whh,
                 const float* __restrict__ bhh, const float* __restrict__ h0,
                 float* __restrict__ yseq, float* __restrict__ hlast) {
  extern __shared__ float smem[];
  float* hbuf  = smem;                 // [32][256]  32 KB
  float* ghbuf = smem + BATCH * HID;   // [32][768]  96 KB

  int tid = threadIdx.x;
#pragma unroll
  for (int e = 0; e < 8; e++) {
    int idx = tid + e * 1024;
    hbuf[idx] = h0 ? h0[idx] : 0.f;
  }
  __syncthreads();

  int w     = tid >> 5;
  int lane  = tid & 31;
  int mlo   = lane & 15;
  int khalf = (lane >> 4) << 1;
  int hi8   = (lane >> 4) << 3;

  for (int t = 0; t < SEQ; t++) {
    // -------- phase 1: gh = h @ W_hh^T (WMMA, A from LDS, B from L2-hot weights) --------
#pragma unroll
    for (int rep = 0; rep < 3; rep++) {
      int flat = w + rep * 32;                  // 0..95, uniform per wave
      int tm = flat / 48, tn = flat % 48;
      const float* hrow = hbuf + (tm * 16 + mlo) * HID + khalf;
      const float* wr   = Whh + (tn * 16 + mlo) * HID + khalf;
      v8f acc = {};
#pragma unroll 2
      for (int kc = 0; kc < HID; kc += 16) {
        v2f a[4], b[4];
#pragma unroll
        for (int s = 0; s < 4; s++) {
          int k = kc + 4 * s;
          a[s].x = hrow[k]; a[s].y = hrow[k + 1];
          b[s].x = wr[k];   b[s].y = wr[k + 1];
        }
#pragma unroll
        for (int s = 0; s < 4; s++)
          acc = wmma4(a[s], b[s], acc);
      }
#pragma unroll
      for (int i = 0; i < 8; i++)
        ghbuf[(tm * 16 + i + hi8) * G3 + tn * 16 + mlo] = acc[i];
    }
    __syncthreads();

    // -------- phase 2: gates + state update --------
#pragma unroll
    for (int e = 0; e < 8; e++) {
      int idx = tid + e * 1024;
      int b = idx >> 8, j = idx & 255;
      const float* xwp = xW + (b * SEQ + t) * G3;
      float xr = xwp[j], xz = xwp[HID + j], xn = xwp[2 * HID + j];
      float hr = ghbuf[b * G3 + j]           + bhh[j];
      float hz = ghbuf[b * G3 + HID + j]     + bhh[HID + j];
      float hn = ghbuf[b * G3 + 2 * HID + j] + bhh[2 * HID + j];
      float r  = sigmoidf_(xr + hr);
      float z  = sigmoidf_(xz + hz);
      float nn = tanhf(xn + r * hn);
      float hnew = (1.f - z) * nn + z * hbuf[b * HID + j];
      hbuf[b * HID + j] = hnew;
      yseq[(b * SEQ + t) * HID + j] = hnew;
    }
    __syncthreads();
  }

#pragma unroll
  for (int e = 0; e < 8; e++) {
    int idx = tid + e * 1024;
    hlast[idx] = hbuf[idx];
  }
}

extern "C" void kernel_launch(void* const* d_in, const int* in_sizes, int n_in,
                              void* d_out, int out_size, void* d_ws, size_t ws_size,
                              hipStream_t stream) {
  const int*   x_sku      = (const int*)d_in[0];
  const int*   x_cat      = (const int*)d_in[1];
  const float* emb_sku    = (const float*)d_in[2];
  const float* emb_cat    = (const float*)d_in[3];
  const float* enc_W_ih   = (const float*)d_in[4];
  const float* enc_W_hh   = (const float*)d_in[5];
  const float* enc_b_ih   = (const float*)d_in[6];
  const float* enc_b_hh   = (const float*)d_in[7];
  const float* dec_W_ih   = (const float*)d_in[8];
  const float* dec_W_hh   = (const float*)d_in[9];
  const float* dec_b_ih   = (const float*)d_in[10];
  const float* dec_b_hh   = (const float*)d_in[11];
  const float* proj_sku_W = (const float*)d_in[12];
  const float* proj_sku_b = (const float*)d_in[13];
  const float* proj_cat_W = (const float*)d_in[14];
  const float* proj_cat_b = (const float*)d_in[15];
  float* out = (float*)d_out;

  float* ws    = (float*)d_ws;
  float* emb   = ws;                         // TOK*HID
  float* yA    = emb + (size_t)TOK * HID;    // TOK*HID
  float* yB    = yA  + (size_t)TOK * HID;    // TOK*HID
  float* xW    = yB  + (size_t)TOK * HID;    // TOK*G3
  float* hE0   = xW  + (size_t)TOK * G3;     // BATCH*HID
  float* hE1   = hE0 + BATCH * HID;          // BATCH*HID
  float* hDump = hE1 + BATCH * HID;          // BATCH*HID

  int nElem = TOK * HID;
  embed_kernel<<<(nElem + 255) / 256, 256, 0, stream>>>(x_sku, x_cat, emb_sku, emb_cat, emb);

  auto gemm = [&](const float* X, const float* W, const float* b, float* Y, int M, int N, bool nt) {
    int nGroups = (N + 63) / 64;
    int waves = (M >> 4) * nGroups;
    int blocks = (waves * 32 + 255) / 256;
    if (nt) gemm_wmma_f32<true ><<<blocks, 256, 0, stream>>>(X, W, b, Y, M, N, nGroups);
    else    gemm_wmma_f32<false><<<blocks, 256, 0, stream>>>(X, W, b, Y, M, N, nGroups);
  };

  size_t smem = (size_t)(BATCH * HID + BATCH * G3) * sizeof(float);  // 128 KB < 320 KB/WGP

  // encoder layer 0
  gemm(emb, enc_W_ih, enc_b_ih, xW, TOK, G3, false);
  gru_recur_kernel<<<1, 1024, smem, stream>>>(xW, enc_W_hh, enc_b_hh, nullptr, yA, hE0);
  // encoder layer 1  (sequence output unused; only hidden matters)
  gemm(yA, enc_W_ih + (size_t)G3 * HID, enc_b_ih + G3, xW, TOK, G3, false);
  gru_recur_kernel<<<1, 1024, smem, stream>>>(xW, enc_W_hh + (size_t)G3 * HID, enc_b_hh + G3,
                                              nullptr, yB, hE1);
  // combined = emb + encoded
  combine_kernel<<<(nElem + 255) / 256, 256, 0, stream>>>(emb, hE1, yA);
  // decoder layer 0 (h0 = encoder layer-0 last hidden)
  gemm(yA, dec_W_ih, dec_b_ih, xW, TOK, G3, false);
  gru_recur_kernel<<<1, 1024, smem, stream>>>(xW, dec_W_hh, dec_b_hh, hE0, yB, hDump);
  // decoder layer 1 (h0 = encoder layer-1 last hidden)
  gemm(yB, dec_W_ih + (size_t)G3 * HID, dec_b_ih + G3, xW, TOK, G3, false);
  gru_recur_kernel<<<1, 1024, smem, stream>>>(xW, dec_W_hh + (size_t)G3 * HID, dec_b_hh + G3,
                                              hE1, yA, hDump);
  // output projections (the 65-GFLOP, 508-MB-store part; weights L2-resident, logits streamed NT)
  gemm(yA, proj_sku_W, proj_sku_b, out, TOK, 30000, true);
  gemm(yA, proj_cat_W, proj_cat_b, out + (size_t)TOK * 30000, TOK, 1000, true);
}